// _GraphConvKerasModel_14834817040497
// MI455X (gfx1250) — compile-verified
//
#include <hip/hip_runtime.h>
#include <hip/hip_bf16.h>
#include <math.h>

#define NTOT   65536
#define GC     256
#define DENSE  512
#define BATCH  2048
#define KP1    76      // 75 padded to multiple of 4
#define BN_EPS 0.001f

typedef float v2f __attribute__((ext_vector_type(2)));
typedef float v8f __attribute__((ext_vector_type(8)));

__constant__ int c_offs[12] = {0, 1536, 9728, 26112, 50688, 62976,
                               64000, 64512, 65024, 65280, 65408, 65536};

struct Adj { const int* p[10]; };

__device__ __forceinline__ int degree_of(int row) {
  int d = 10;
  while (row < c_offs[d]) --d;
  return d;
}

// ---------------- weight repack: row-major (Ksrc x C) -> k4-fragment-major ----
// dst[m][ (k>>2)*C + c ][k&3]  so a WMMA B-fragment (2 consecutive k at one col)
// is a single aligned b64 load, coalesced across lanes. Zero-pads k >= Ksrc.

__global__ void repack_w_kernel(const float* __restrict__ src, float* __restrict__ dst,
                                int M, int Ksrc, int K, int C) {
  int tid = blockIdx.x * blockDim.x + threadIdx.x;   // grid = M*K*C/256 exact
  int m = tid / (K * C);
  int rem = tid - m * (K * C);
  int k = rem / C, c = rem - k * C;
  float v = (k < Ksrc) ? src[(size_t)m * Ksrc * C + (size_t)k * C + c] : 0.0f;
  dst[(size_t)m * K * C + ((size_t)(k >> 2) * C + c) * 4 + (k & 3)] = v;
}

// ---------------- activation pad (75 -> 76 cols) ----------------

__global__ void pad_x_kernel(const float* __restrict__ A, float* __restrict__ XP) {
  int tid = blockIdx.x * blockDim.x + threadIdx.x;   // N*76 exact
  int r = tid / KP1, c = tid - r * KP1;
  XP[tid] = (c < 75) ? A[(size_t)r * 75 + c] : 0.0f;
}

// ---------------- neighbor gather-sum (rows >= OFFS[1]) ----------------

__global__ void gather_sum_kernel(const float* __restrict__ X, float* __restrict__ O,
                                  Adj adj, int ng /* = KP/4 float4 groups */) {
  int tid = blockIdx.x * blockDim.x + threadIdx.x;
  int row_off = tid / ng;
  int cg = tid - row_off * ng;
  if (row_off >= NTOT - 1536) return;
  int row = 1536 + row_off;
  int d = degree_of(row);
  const int* a = adj.p[d - 1] + (size_t)(row - c_offs[d]) * d;
  const float4* x4 = (const float4*)X;
  float4 s = make_float4(0.f, 0.f, 0.f, 0.f);
  for (int j = 0; j < d; ++j) {
    float4 nv = x4[(size_t)a[j] * ng + cg];
    s.x += nv.x; s.y += nv.y; s.z += nv.z; s.w += nv.w;
  }
  ((float4*)O)[(size_t)row * ng + cg] = s;
}

// ---------------- graph-conv GEMM via fp32 WMMA ----------------
// 8 waves / 256 threads; block tile = 64 rows x 256 cols; wave = 16 rows x 128 cols.
// Degree derived from blockIdx only => pure scalar branches, EXEC stays all-1s.
// Self-activation tile staged in LDS (stride KPS, gcd(KPS,64)<=4 => conflict-free
// b64 fragment reads). Weights come from k4-repacked layout (b64, coalesced).

template <int KP, int KPS>
__global__ __launch_bounds__(256)
void gc_gemm_kernel(const float* __restrict__ X, const float* __restrict__ NG,
                    const float* __restrict__ W0, const float* __restrict__ WS,
                    const float* __restrict__ WN, const float* __restrict__ BIAS,
                    const float* __restrict__ BG, const float* __restrict__ BB,
                    const float* __restrict__ BM, const float* __restrict__ BV,
                    float* __restrict__ OUT) {
  extern __shared__ float sX[];                 // 64*KPS floats
  const int tid  = threadIdx.x;
  const int lane = tid & 31;
  const int wid  = tid >> 5;
  const int l16  = lane & 15;
  const int half = lane >> 4;
  const int rblk = blockIdx.x * 64;
  const int d    = degree_of(rblk);             // uniform (SGPR) by construction

  // stage 64 x KP self-activation tile (contiguous rows => linear, coalesced)
  {
    const float4* g = (const float4*)(X + (size_t)rblk * KP);
    for (int i = tid; i < 64 * (KP / 4); i += 256) {
      int r = i / (KP / 4), cq = i - r * (KP / 4);
      *(float4*)(sX + r * KPS + cq * 4) = g[i];
    }
  }
  __syncthreads();

  const int dm1 = (d > 0) ? (d - 1) : 0;
  const float* WA = (d == 0) ? W0 : (WS + (size_t)dm1 * KP * GC);
  const float* WB = WN + (size_t)dm1 * KP * GC;

  v8f acc[8];
#pragma unroll
  for (int t = 0; t < 8; ++t)
#pragma unroll
    for (int v = 0; v < 8; ++v) acc[t][v] = 0.0f;

  const int cb = (wid & 1) * 128;
  const float* sxr  = sX + ((wid >> 1) * 16 + l16) * KPS;
  const float* nrow = NG + (size_t)(rblk + (wid >> 1) * 16 + l16) * KP;

#pragma unroll 2
  for (int kt = 0; kt < KP / 4; ++kt) {
    const int ka = kt * 4 + half * 2;
    v2f aS = *(const v2f*)(sxr + ka);
    v2f bS[8];
#pragma unroll
    for (int t = 0; t < 8; ++t)
      bS[t] = *(const v2f*)(WA + ((size_t)kt * GC + cb + t * 16 + l16) * 4 + half * 2);
    if (d > 0) {
      v2f aN = *(const v2f*)(nrow + ka);
      v2f bN[8];
#pragma unroll
      for (int t = 0; t < 8; ++t)
        bN[t] = *(const v2f*)(WB + ((size_t)kt * GC + cb + t * 16 + l16) * 4 + half * 2);
#pragma unroll
      for (int t = 0; t < 8; ++t) {
        acc[t] = __builtin_amdgcn_wmma_f32_16x16x4_f32(
            false, aS, false, bS[t], (short)0, acc[t], false, false);
        acc[t] = __builtin_amdgcn_wmma_f32_16x16x4_f32(
            false, aN, false, bN[t], (short)0, acc[t], false, false);
      }
    } else {
#pragma unroll
      for (int t = 0; t < 8; ++t)
        acc[t] = __builtin_amdgcn_wmma_f32_16x16x4_f32(
            false, aS, false, bS[t], (short)0, acc[t], false, false);
    }
  }

  const float* bias = BIAS + d * GC;
#pragma unroll
  for (int t = 0; t < 8; ++t) {
    const int c = cb + t * 16 + l16;
    const float g = BG[c], be = BB[c], mu = BM[c];
    const float inv = rsqrtf(BV[c] + BN_EPS);
    const float bs = bias[c];
#pragma unroll
    for (int v = 0; v < 8; ++v) {
      const int row = rblk + (wid >> 1) * 16 + v + half * 8;
      float val = acc[t][v] + bs;
      val = fmaxf(val, 0.0f);                 // relu BEFORE bn (reference order)
      val = g * (val - mu) * inv + be;
      OUT[(size_t)row * GC + c] = val;
    }
  }
}

// ---------------- graph max-pool ----------------

__global__ void pool_kernel(const float* __restrict__ H, float* __restrict__ O, Adj adj) {
  int tid = blockIdx.x * blockDim.x + threadIdx.x;   // N*64 exact
  int row = tid >> 6;
  int cg  = tid & 63;
  const float4* h4 = (const float4*)H;
  float4 v = h4[(size_t)row * 64 + cg];
  int d = degree_of(row);
  if (d > 0) {
    const int* a = adj.p[d - 1] + (size_t)(row - c_offs[d]) * d;
    for (int j = 0; j < d; ++j) {
      float4 nv = h4[(size_t)a[j] * 64 + cg];
      v.x = fmaxf(v.x, nv.x); v.y = fmaxf(v.y, nv.y);
      v.z = fmaxf(v.z, nv.z); v.w = fmaxf(v.w, nv.w);
    }
  }
  ((float4*)O)[(size_t)row * 64 + cg] = v;
}

// ---------------- dense layer (256 -> 512) via fp32 WMMA ----------------
// 8 waves; block tile = 32 rows x 512 cols; wave: strip=(w>>2)*16, cols (w&3)*128.

__global__ __launch_bounds__(256)
void dense_gemm_kernel(const float* __restrict__ X, const float* __restrict__ W,
                       const float* __restrict__ B,  const float* __restrict__ BG,
                       const float* __restrict__ BB, const float* __restrict__ BM,
                       const float* __restrict__ BV, float* __restrict__ OUT) {
  extern __shared__ float sX[];                 // 32*260 floats
  const int KPS = 260;
  const int tid  = threadIdx.x;
  const int lane = tid & 31;
  const int wid  = tid >> 5;
  const int l16  = lane & 15;
  const int half = lane >> 4;
  const int rblk = blockIdx.x * 32;

  {
    const float4* g = (const float4*)(X + (size_t)rblk * GC);
    for (int i = tid; i < 32 * (GC / 4); i += 256) {
      int r = i / (GC / 4), cq = i - r * (GC / 4);
      *(float4*)(sX + r * KPS + cq * 4) = g[i];
    }
  }
  __syncthreads();

  v8f acc[8];
#pragma unroll
  for (int t = 0; t < 8; ++t)
#pragma unroll
    for (int v = 0; v < 8; ++v) acc[t][v] = 0.0f;

  const int cb = (wid & 3) * 128;
  const float* sxr = sX + ((wid >> 2) * 16 + l16) * KPS;

#pragma unroll 2
  for (int kt = 0; kt < GC / 4; ++kt) {
    const int ka = kt * 4 + half * 2;
    v2f a = *(const v2f*)(sxr + ka);
    v2f b[8];
#pragma unroll
    for (int t = 0; t < 8; ++t)
      b[t] = *(const v2f*)(W + ((size_t)kt * DENSE + cb + t * 16 + l16) * 4 + half * 2);
#pragma unroll
    for (int t = 0; t < 8; ++t)
      acc[t] = __builtin_amdgcn_wmma_f32_16x16x4_f32(
          false, a, false, b[t], (short)0, acc[t], false, false);
  }

#pragma unroll
  for (int t = 0; t < 8; ++t) {
    const int c = cb + t * 16 + l16;
    const float g = BG[c], be = BB[c], mu = BM[c];
    const float inv = rsqrtf(BV[c] + BN_EPS);
    const float bs = B[c];
#pragma unroll
    for (int v = 0; v < 8; ++v) {
      const int row = rblk + (wid >> 2) * 16 + v + half * 8;
      float val = acc[t][v] + bs;
      val = fmaxf(val, 0.0f);
      val = g * (val - mu) * inv + be;
      OUT[(size_t)row * DENSE + c] = val;
    }
  }
}

// ---------------- segment bucketing + reduce + tanh fingerprint ----------------

__global__ void seg_init_kernel(int* cnt) {
  int t = blockIdx.x * blockDim.x + threadIdx.x;
  if (t < BATCH) cnt[t] = 0;
}

__global__ void seg_build_kernel(const int* __restrict__ mem, int* cnt, int* idx) {
  int r = blockIdx.x * blockDim.x + threadIdx.x;   // N exact
  int s = mem[r];
  int p = atomicAdd(&cnt[s], 1);
  idx[s * 32 + p] = r;                             // each segment has exactly 32
}

__global__ void seg_reduce_kernel(const float* __restrict__ D,
                                  const int* __restrict__ idx,
                                  float* __restrict__ fp) {
  int s = blockIdx.x;            // one block per segment
  int c0 = threadIdx.x;          // 256 threads, 2 cols each
  for (int u = 0; u < 2; ++u) {
    int c = c0 + u * 256;
    float sum = 0.0f, mx = -3.402823466e38f;
    for (int j = 0; j < 32; ++j) {
      int r = idx[s * 32 + j];
      float v = D[(size_t)r * DENSE + c];
      sum += v;
      mx = fmaxf(mx, v);
    }
    fp[(size_t)s * 1024 + c]       = tanhf(sum);   // cols 0..511: segment sum
    fp[(size_t)s * 1024 + 512 + c] = tanhf(mx);    // cols 512..1023: segment max
  }
}

// ---------------- output GEMM + softmax (2048 x 1024 @ 1024 x 24) ----------------

__global__ void out_kernel(const float* __restrict__ fp, const float* __restrict__ W,
                           const float* __restrict__ B, float* __restrict__ probs,
                           float* __restrict__ logits) {
  int t = blockIdx.x * blockDim.x + threadIdx.x;   // 2048*12 exact
  int s = t / 12, k = t - s * 12;
  const float* f = fp + (size_t)s * 1024;
  float l0 = B[k * 2], l1 = B[k * 2 + 1];
  for (int i = 0; i < 1024; ++i) {
    float fv = f[i];
    l0 += fv * W[(size_t)i * 24 + k * 2];
    l1 += fv * W[(size_t)i * 24 + k * 2 + 1];
  }
  size_t o = (size_t)s * 24 + k * 2;
  logits[o] = l0; logits[o + 1] = l1;
  float m = fmaxf(l0, l1);
  float e0 = expf(l0 - m), e1 = expf(l1 - m);
  float inv = 1.0f / (e0 + e1);
  probs[o] = e0 * inv; probs[o + 1] = e1 * inv;
}

// ---------------- launcher ----------------

extern "C" void kernel_launch(void* const* d_in, const int* in_sizes, int n_in,
                              void* d_out, int out_size, void* d_ws, size_t ws_size,
                              hipStream_t stream) {
  (void)in_sizes; (void)n_in; (void)out_size; (void)ws_size;

  const float* atom       = (const float*)d_in[0];
  const int*   membership = (const int*)d_in[2];
  Adj adj;
  for (int i = 0; i < 10; ++i) adj.p[i] = (const int*)d_in[4 + i];
  const float* gc1_w0     = (const float*)d_in[14];
  const float* gc1_wself  = (const float*)d_in[15];
  const float* gc1_wneigh = (const float*)d_in[16];
  const float* gc1_b      = (const float*)d_in[17];
  const float* gc2_w0     = (const float*)d_in[18];
  const float* gc2_wself  = (const float*)d_in[19];
  const float* gc2_wneigh = (const float*)d_in[20];
  const float* gc2_b      = (const float*)d_in[21];
  const float* bn1g = (const float*)d_in[22], *bn1b = (const float*)d_in[23];
  const float* bn1m = (const float*)d_in[24], *bn1v = (const float*)d_in[25];
  const float* bn2g = (const float*)d_in[26], *bn2b = (const float*)d_in[27];
  const float* bn2m = (const float*)d_in[28], *bn2v = (const float*)d_in[29];
  const float* bn3g = (const float*)d_in[30], *bn3b = (const float*)d_in[31];
  const float* bn3m = (const float*)d_in[32], *bn3v = (const float*)d_in[33];
  const float* dense_w = (const float*)d_in[34];
  const float* dense_b = (const float*)d_in[35];
  const float* out_w   = (const float*)d_in[36];
  const float* out_b   = (const float*)d_in[37];

  float* ws = (float*)d_ws;
  // region-reused workspace layout (float offsets); peak ~250 MB
  const size_t WP1_OFF = 0;                                     // 21*76*256
  const size_t WP2_OFF = WP1_OFF + (size_t)21 * KP1 * GC;       // 21*256*256
  const size_t WPD_OFF = WP2_OFF + (size_t)21 * GC * GC;        // 256*512
  const size_t XP_OFF  = WPD_OFF + (size_t)GC * DENSE;          // N*76
  const size_t NG1_OFF = XP_OFF  + (size_t)NTOT * KP1;          // N*76
  const size_t H1_OFF  = NG1_OFF + (size_t)NTOT * KP1;          // N*256
  const size_t H1P_OFF = H1_OFF  + (size_t)NTOT * GC;           // N*256
  const size_t NG2_OFF = H1P_OFF + (size_t)NTOT * GC;           // N*256
  const size_t H2_OFF  = H1_OFF;     // reuse: H1 dead after pool1
  const size_t H2P_OFF = NG2_OFF;    // reuse: NG2 dead after gc2 GEMM
  const size_t D_OFF   = XP_OFF;     // reuse: XP/NG1/H2/H1P dead at dense time
  const size_t INT_OFF = NG2_OFF + (size_t)NTOT * GC;           // ints after peak

  float* WP1 = ws + WP1_OFF;
  float* WP2 = ws + WP2_OFF;
  float* WPD = ws + WPD_OFF;
  float* XP  = ws + XP_OFF;
  float* NG1 = ws + NG1_OFF;
  float* H1  = ws + H1_OFF;
  float* H1P = ws + H1P_OFF;
  float* NG2 = ws + NG2_OFF;
  float* H2  = ws + H2_OFF;
  float* H2P = ws + H2P_OFF;
  float* Dm  = ws + D_OFF;
  int* cnt = (int*)(ws + INT_OFF);
  int* idx = cnt + BATCH;

  float* probs  = (float*)d_out;                 // 2048*24
  float* logits = probs + (size_t)BATCH * 24;    // 2048*24
  float* fp     = logits + (size_t)BATCH * 24;   // 2048*1024

  // ---- weight repacks (k4 fragment-major, zero-padded K) ----
  repack_w_kernel<<<(1 * KP1 * GC) / 256, 256, 0, stream>>>(gc1_w0, WP1, 1, 75, KP1, GC);
  repack_w_kernel<<<(10 * KP1 * GC) / 256, 256, 0, stream>>>(gc1_wself, WP1 + (size_t)KP1 * GC, 10, 75, KP1, GC);
  repack_w_kernel<<<(10 * KP1 * GC) / 256, 256, 0, stream>>>(gc1_wneigh, WP1 + (size_t)11 * KP1 * GC, 10, 75, KP1, GC);
  repack_w_kernel<<<(1 * GC * GC) / 256, 256, 0, stream>>>(gc2_w0, WP2, 1, GC, GC, GC);
  repack_w_kernel<<<(10 * GC * GC) / 256, 256, 0, stream>>>(gc2_wself, WP2 + (size_t)GC * GC, 10, GC, GC, GC);
  repack_w_kernel<<<(10 * GC * GC) / 256, 256, 0, stream>>>(gc2_wneigh, WP2 + (size_t)11 * GC * GC, 10, GC, GC, GC);
  repack_w_kernel<<<(GC * DENSE) / 256, 256, 0, stream>>>(dense_w, WPD, 1, GC, GC, DENSE);

  // ---- layer 1 ----
  pad_x_kernel<<<(NTOT * KP1) / 256, 256, 0, stream>>>(atom, XP);
  gather_sum_kernel<<<((NTOT - 1536) * (KP1 / 4)) / 256, 256, 0, stream>>>(XP, NG1, adj, KP1 / 4);
  gc_gemm_kernel<KP1, 84><<<NTOT / 64, 256, 64 * 84 * 4, stream>>>(
      XP, NG1, WP1, WP1 + (size_t)KP1 * GC, WP1 + (size_t)11 * KP1 * GC,
      gc1_b, bn1g, bn1b, bn1m, bn1v, H1);
  pool_kernel<<<(NTOT * 64) / 256, 256, 0, stream>>>(H1, H1P, adj);

  // ---- layer 2 ----
  gather_sum_kernel<<<((NTOT - 1536) * (GC / 4)) / 256, 256, 0, stream>>>(H1P, NG2, adj, GC / 4);
  gc_gemm_kernel<GC, 260><<<NTOT / 64, 256, 64 * 260 * 4, stream>>>(
      H1P, NG2, WP2, WP2 + (size_t)GC * GC, WP2 + (size_t)11 * GC * GC,
      gc2_b, bn2g, bn2b, bn2m, bn2v, H2);
  pool_kernel<<<(NTOT * 64) / 256, 256, 0, stream>>>(H2, H2P, adj);

  // ---- dense + bn3 ----
  dense_gemm_kernel<<<NTOT / 32, 256, 32 * 260 * 4, stream>>>(
      H2P, WPD, dense_b, bn3g, bn3b, bn3m, bn3v, Dm);

  // ---- segment sum/max -> tanh fingerprint ----
  seg_init_kernel<<<BATCH / 256, 256, 0, stream>>>(cnt);
  seg_build_kernel<<<NTOT / 256, 256, 0, stream>>>(membership, cnt, idx);
  seg_reduce_kernel<<<BATCH, 256, 0, stream>>>(Dm, idx, fp);

  // ---- output head ----
  out_kernel<<<(BATCH * 12) / 256, 256, 0, stream>>>(fp, out_w, out_b, probs, logits);
}